// MultiHeadAttentionWithCache_13709535609483
// MI455X (gfx1250) — compile-verified
//
#include <hip/hip_runtime.h>
#include <math.h>

#define EMBED 4096
#define HEADS 32
#define HD 128
#define KV 8192
#define CHUNK 1024
#define NCHUNK (KV / CHUNK)              // 8
#define NT (CHUNK / 16)                  // 64 position-tiles per chunk
#define CACHE_ELEMS ((size_t)KV * EMBED) // 33554432
#define CACHE_F4 (CACHE_ELEMS / 4)       // 8388608
#define POS_F4 (EMBED / 4)               // 1024 float4 per cache position
#define ROW_STRIDE 132                   // 128 + 4 pad DWORDs (bank-conflict-free)

typedef __attribute__((ext_vector_type(2))) float v2f;
typedef __attribute__((ext_vector_type(8))) float v8f;
typedef __attribute__((ext_vector_type(4))) unsigned int u32x4;
typedef __attribute__((ext_vector_type(4))) int i32x4;
typedef __attribute__((ext_vector_type(8))) int i32x8;

#if __has_builtin(__builtin_amdgcn_tensor_load_to_lds) && __has_builtin(__builtin_amdgcn_s_wait_tensorcnt)
#define USE_TDM 1
#else
#define USE_TDM 0
#endif

// workspace layout (float units)
#define WS_Q    0
#define WS_K    4096
#define WS_V    8192
#define WS_VAL  12288
#define WS_MASK 16384          // int[8192]
#define WS_PART 24576          // 32 heads * 8 chunks * 130 floats
#define PART_STRIDE 130

// ---------------------------------------------------------------------------
// Init: zero mask, seed GEMV accumulators with biases.
// ---------------------------------------------------------------------------
__global__ void init_kernel(float* __restrict__ ws, float* __restrict__ out,
                            const float* __restrict__ bq, const float* __restrict__ bk,
                            const float* __restrict__ bv, const float* __restrict__ bo) {
  int id = blockIdx.x * blockDim.x + threadIdx.x;
  if (id < KV) {
    ((int*)(ws + WS_MASK))[id] = 0;
  }
  int r = id - KV;
  if (r >= 0 && r < 4 * EMBED) {
    int seg = r / EMBED, e = r % EMBED;
    if (seg == 0)      ws[WS_Q + e] = bq[e];
    else if (seg == 1) ws[WS_K + e] = bk[e];
    else if (seg == 2) ws[WS_V + e] = bv[e];
    else               out[e] = bo[e];
  }
}

// ---------------------------------------------------------------------------
// GEMV: acc[j] += sum_i x[i] * W[i][j], split over row chunks (atomicAdd).
// ---------------------------------------------------------------------------
__global__ void gemv4096_kernel(const float* __restrict__ x, const float* __restrict__ W,
                                float* __restrict__ acc) {
  const int j4 = blockIdx.x * blockDim.x + threadIdx.x;  // float4 column index
  const int i0 = blockIdx.y * 128;
  const float4* __restrict__ W4 = (const float4*)W;
  float s0 = 0.f, s1 = 0.f, s2 = 0.f, s3 = 0.f;
#pragma unroll 8
  for (int i = i0; i < i0 + 128; ++i) {
    const float xv = __ldg(&x[i]);
    const float4 w = W4[(size_t)i * (EMBED / 4) + j4];
    s0 += xv * w.x; s1 += xv * w.y; s2 += xv * w.z; s3 += xv * w.w;
  }
  atomicAdd(&acc[j4 * 4 + 0], s0);
  atomicAdd(&acc[j4 * 4 + 1], s1);
  atomicAdd(&acc[j4 * 4 + 2], s2);
  atomicAdd(&acc[j4 * 4 + 3], s3);
}

// ---------------------------------------------------------------------------
// Roll both caches into d_out and compute slot mask via ballot + atomicOr.
// ---------------------------------------------------------------------------
__global__ void roll_mask_kernel(const float* __restrict__ vin, const float* __restrict__ kin,
                                 const float* __restrict__ vnew, const float* __restrict__ knew,
                                 float* __restrict__ out, int* __restrict__ mask) {
  const size_t stride = (size_t)gridDim.x * blockDim.x;
  float4* __restrict__ outv = (float4*)(out + EMBED);
  float4* __restrict__ outk = (float4*)(out + EMBED + CACHE_ELEMS);
  const float4* __restrict__ v4 = (const float4*)vin;
  const float4* __restrict__ k4 = (const float4*)kin;
  const float4* __restrict__ vn4 = (const float4*)vnew;
  const float4* __restrict__ kn4 = (const float4*)knew;
  const int lane = threadIdx.x & 31;
  for (size_t idx = (size_t)blockIdx.x * blockDim.x + threadIdx.x;
       idx < 2ull * CACHE_F4; idx += stride) {
    const bool is_k = idx >= CACHE_F4;              // wave-uniform (CACHE_F4 % 32 == 0)
    const size_t r = is_k ? idx - CACHE_F4 : idx;
    float4 val;
    if (r < CACHE_F4 - POS_F4)
      val = is_k ? k4[r + POS_F4] : v4[r + POS_F4];
    else
      val = is_k ? kn4[r - (CACHE_F4 - POS_F4)] : vn4[r - (CACHE_F4 - POS_F4)];
    if (is_k) outk[r] = val; else outv[r] = val;
    if (!is_k) {
      const bool nz = (val.x != 0.f) | (val.y != 0.f) | (val.z != 0.f) | (val.w != 0.f);
      const unsigned long long b = __ballot(nz);    // wave stays within one position
      if (lane == 0 && b != 0ull) atomicOr(&mask[(int)(r / POS_F4)], 1);
    }
  }
}

// ---------------------------------------------------------------------------
// Split-KV flash-decode attention partial. One block = (head, KV chunk).
// K tiles are staged into LDS by the Tensor Data Mover (double buffered,
// TENSORcnt-synchronized); scores via V_WMMA_F32_16X16X4_F32 with q
// replicated into all 16 A rows (held in registers). V read in float4 clauses.
// ---------------------------------------------------------------------------
struct AttnSmem {
  float kstage[2][8][16 * ROW_STRIDE]; // 2 bufs x 8 waves x padded 16x128 tile
  float q[HD];
  float red_m[8], red_l[8];
  float red_acc[8][HD];
};

__global__ void attn_partial_kernel(const float* __restrict__ ws, const float* __restrict__ out,
                                    float* __restrict__ partials) {
  const int h = blockIdx.y;
  const int chunk = blockIdx.x;
  const int tid = threadIdx.x;
  const int wave = tid >> 5;
  const int lane = tid & 31;
  const int half = lane >> 4;   // 0: K-cols {0,1}, 1: K-cols {2,3}
  const int nidx = lane & 15;   // position within 16-wide tile
  const float* __restrict__ V = out + EMBED;
  const float* __restrict__ K = out + EMBED + CACHE_ELEMS;
  const int* __restrict__ mask = (const int*)(ws + WS_MASK);

  __shared__ AttnSmem sm;

  if (tid < HD) sm.q[tid] = ws[WS_Q + h * HD + tid];
  __syncthreads();

  // Hoist q (A operand pairs, replicated rows) into registers: loop-invariant.
  v2f aq[32];
#pragma unroll
  for (int kk = 0; kk < 32; ++kk) {
    aq[kk][0] = sm.q[kk * 4 + half * 2];
    aq[kk][1] = sm.q[kk * 4 + half * 2 + 1];
  }

  float m = -INFINITY, l = 0.f;
  float a0 = 0.f, a1 = 0.f, a2 = 0.f, a3 = 0.f;
  const float scale = 0.08838834764831845f;  // 1/sqrt(128)
  const int p_base = chunk * CHUNK;

#if USE_TDM
  // Issue one TDM descriptor: 16 rows x 128 f32, row stride 4096 elems,
  // LDS-padded to ROW_STRIDE dwords (pad_interval=128DW, pad_amount=4DW).
  auto issue_tile = [&](int t, int buf) {
    const float* gtile = K + (size_t)(p_base + t * 16) * EMBED + h * HD;
    const unsigned lds_off = (unsigned)(size_t)(void*)&sm.kstage[buf][wave][0];
    const unsigned long long ga = (unsigned long long)(size_t)gtile;
    u32x4 g0 = { 1u,                                    // count=1, user descriptor
                 lds_off,                               // lds_addr
                 (unsigned)(ga & 0xFFFFFFFFu),          // global_addr[31:0]
                 (unsigned)((ga >> 32) & 0x01FFFFFFu) | (2u << 30) }; // [56:32] | type=2
    i32x8 g1 = { (int)((2u << 16) | (1u << 20) | (6u << 22) | (3u << 25)),
                 // data_size=4B, pad_enable, pad_interval=128DW, pad_amount=4DW
                 (int)(128u << 16),   // tensor_dim0[15:0] << 16
                 (int)(16u << 16),    // tensor_dim0[31:16]=0 | tensor_dim1[15:0]<<16
                 (int)(128u << 16),   // tensor_dim1[31:16]=0 | tile_dim0<<16
                 16,                  // tile_dim1=16, tile_dim2=0
                 4096,                // tensor_dim0_stride[31:0]
                 0, 0 };
    i32x4 gz = { 0, 0, 0, 0 };
#if __clang_major__ >= 23
    i32x8 gz8 = { 0, 0, 0, 0, 0, 0, 0, 0 };
    __builtin_amdgcn_tensor_load_to_lds(g0, g1, gz, gz, gz8, 0);
#else
    __builtin_amdgcn_tensor_load_to_lds(g0, g1, gz, gz, 0);
#endif
  };
  issue_tile(wave, 0);
#endif

  int buf = 0;
  for (int t = wave; t < NT; t += 8) {
    const int p0 = p_base + t * 16;

#if USE_TDM
    if (t + 8 < NT) {
      issue_tile(t + 8, buf ^ 1);             // prefetch next tile via TDM
      __builtin_amdgcn_s_wait_tensorcnt(1);   // current tile has landed
    } else {
      __builtin_amdgcn_s_wait_tensorcnt(0);
    }
    asm volatile("" ::: "memory");            // keep LDS reads below the wait
#endif

    // batched V tile read (16 independent float4 loads -> one clause)
    const float* __restrict__ vbase = V + (size_t)p0 * EMBED + h * HD + lane * 4;
    __builtin_prefetch(vbase + 128ull * EMBED, 0, 3);  // wave's next V tile, near cache
    float4 vv[16];
#pragma unroll
    for (int p = 0; p < 16; ++p) vv[p] = *(const float4*)(vbase + (size_t)p * EMBED);

    // scores: 32 chained WMMAs over d=0..127
    v8f c = {};
#if USE_TDM
    const float* __restrict__ kb = &sm.kstage[buf][wave][0];
#else
    const float* __restrict__ krow = K + (size_t)(p0 + nidx) * EMBED + h * HD + half * 2;
#endif
#pragma unroll
    for (int kk = 0; kk < 32; ++kk) {
      v2f b;
#if USE_TDM
      const float2 kl = *(const float2*)(kb + nidx * ROW_STRIDE + kk * 4 + half * 2);
#else
      const float2 kl = *(const float2*)(krow + kk * 4);
#endif
      b[0] = kl.x; b[1] = kl.y;
      c = __builtin_amdgcn_wmma_f32_16x16x4_f32(false, aq[kk], false, b, (short)0, c, false, false);
    }
    buf ^= 1;

    float s = c[0] * scale;
    if (mask[p0 + nidx] == 0) s = -INFINITY;

    // online softmax over this 16-position tile (group-of-16 reductions)
    float tmax = s;
#pragma unroll
    for (int off = 8; off >= 1; off >>= 1)
      tmax = fmaxf(tmax, __shfl_xor(tmax, off, 32));
    const float m_new = fmaxf(m, tmax);
    const float w = (s == -INFINITY) ? 0.f : __expf(s - m_new);
    const float corr = (m == -INFINITY) ? 0.f : __expf(m - m_new);
    float wsum = w;
#pragma unroll
    for (int off = 8; off >= 1; off >>= 1)
      wsum += __shfl_xor(wsum, off, 32);
    l = l * corr + wsum;
    m = m_new;
    a0 *= corr; a1 *= corr; a2 *= corr; a3 *= corr;

    // attn @ V on registers; lane owns d = lane*4..lane*4+3
#pragma unroll
    for (int p = 0; p < 16; ++p) {
      const float wp = __shfl(w, p, 32);
      a0 += wp * vv[p].x; a1 += wp * vv[p].y; a2 += wp * vv[p].z; a3 += wp * vv[p].w;
    }
  }

  if (lane == 0) { sm.red_m[wave] = m; sm.red_l[wave] = l; }
  sm.red_acc[wave][lane * 4 + 0] = a0;
  sm.red_acc[wave][lane * 4 + 1] = a1;
  sm.red_acc[wave][lane * 4 + 2] = a2;
  sm.red_acc[wave][lane * 4 + 3] = a3;
  __syncthreads();

  if (tid < HD) {
    float M = -INFINITY;
#pragma unroll
    for (int wv = 0; wv < 8; ++wv) M = fmaxf(M, sm.red_m[wv]);
    float L = 0.f, A = 0.f;
#pragma unroll
    for (int wv = 0; wv < 8; ++wv) {
      const float e = (sm.red_m[wv] == -INFINITY) ? 0.f : __expf(sm.red_m[wv] - M);
      L += sm.red_l[wv] * e;
      A += sm.red_acc[wv][tid] * e;
    }
    float* part = partials + (size_t)(h * NCHUNK + chunk) * PART_STRIDE;
    if (tid == 0) { part[0] = M; part[1] = L; }
    part[2 + tid] = A;
  }
}

// ---------------------------------------------------------------------------
// Combine per-chunk partials -> normalized attention output per head.
// ---------------------------------------------------------------------------
__global__ void attn_reduce_kernel(const float* __restrict__ partials,
                                   float* __restrict__ values) {
  const int h = blockIdx.x;
  const int d = threadIdx.x;
  float M = -INFINITY;
#pragma unroll
  for (int c = 0; c < NCHUNK; ++c)
    M = fmaxf(M, partials[(size_t)(h * NCHUNK + c) * PART_STRIDE]);
  float L = 0.f, A = 0.f;
#pragma unroll
  for (int c = 0; c < NCHUNK; ++c) {
    const float* part = partials + (size_t)(h * NCHUNK + c) * PART_STRIDE;
    const float e = (part[0] == -INFINITY) ? 0.f : __expf(part[0] - M);
    L += part[1] * e;
    A += part[2 + d] * e;
  }
  values[h * HD + d] = A / L;
}

// ---------------------------------------------------------------------------
extern "C" void kernel_launch(void* const* d_in, const int* in_sizes, int n_in,
                              void* d_out, int out_size, void* d_ws, size_t ws_size,
                              hipStream_t stream) {
  const float* x  = (const float*)d_in[0];
  const float* v  = (const float*)d_in[1];
  const float* k  = (const float*)d_in[2];
  const float* Wv = (const float*)d_in[3];
  const float* bv = (const float*)d_in[4];
  const float* Wq = (const float*)d_in[5];
  const float* bq = (const float*)d_in[6];
  const float* Wk = (const float*)d_in[7];
  const float* bk = (const float*)d_in[8];
  const float* Wo = (const float*)d_in[9];
  const float* bo = (const float*)d_in[10];
  float* out = (float*)d_out;
  float* ws  = (float*)d_ws;

  init_kernel<<<96, 256, 0, stream>>>(ws, out, bq, bk, bv, bo);

  const dim3 ggrid(EMBED / 1024, EMBED / 128);  // (4, 32)
  gemv4096_kernel<<<ggrid, 256, 0, stream>>>(x, Wq, ws + WS_Q);
  gemv4096_kernel<<<ggrid, 256, 0, stream>>>(x, Wk, ws + WS_K);
  gemv4096_kernel<<<ggrid, 256, 0, stream>>>(x, Wv, ws + WS_V);

  roll_mask_kernel<<<8192, 256, 0, stream>>>(v, k, ws + WS_V, ws + WS_K,
                                             out, (int*)(ws + WS_MASK));

  attn_partial_kernel<<<dim3(NCHUNK, HEADS), 256, 0, stream>>>(ws, out, ws + WS_PART);
  attn_reduce_kernel<<<HEADS, HD, 0, stream>>>(ws + WS_PART, ws + WS_VAL);

  gemv4096_kernel<<<ggrid, 256, 0, stream>>>(ws + WS_VAL, Wo, out);
}